// BaseDepthTransform_63943473103193
// MI455X (gfx1250) — compile-verified
//
#include <hip/hip_runtime.h>
#include <cstdint>
#include <cstddef>

#define NCAM 6
#define DD   59
#define FHH  32
#define FWW  88
#define CDIM 80
#define NX   360
#define NY   360
#define BEV_ELEMS (CDIM*NX*NY)   /* 10,368,000 */
#define IHH  256
#define IWW  704
#define PIX  (IHH*IWW)           /* 180,224 */
#define NPT  30000
#define PE   (NPT*8)             /* 240,000 */

typedef unsigned int u32x4 __attribute__((ext_vector_type(4)));
typedef int          i32x4 __attribute__((ext_vector_type(4)));
typedef int          i32x8 __attribute__((ext_vector_type(8)));

#if defined(__has_builtin)
#  if __has_builtin(__builtin_amdgcn_tensor_load_to_lds)
#    define HAS_TDM 1
#  endif
#  if __has_builtin(__builtin_amdgcn_s_wait_tensorcnt)
#    define HAS_TWAIT 1
#  endif
#endif

// ------------------------------------------------------------- BEV pooling
// One block per (cam, depth-bin, image-column w). The TDM pulls the 32x80
// f32 tile (row stride FWW*CDIM elems, i.e. 32 rows of 320B stride 28160B)
// into LDS as one DMA; threads 0..31 compute the voxel index per image-row
// h; threads 0..79 run-length aggregate over h (the h axis collapses into
// the single z bin, so consecutive h share BEV cells) and emit
// global_atomic_add_f32 into the BEV grid (layout [c][x][y]).
__global__ __launch_bounds__(96) void bev_pool_kernel(
    const float* __restrict__ feats,
    const float* __restrict__ P,
    float* __restrict__ bev) {
  __shared__ __align__(16) float tile[FHH * CDIM];
  __shared__ int sidx[FHH];

  const int blk = blockIdx.x;
  const int w   = blk % FWW;
  const int dn  = blk / FWW;
  const int d   = dn % DD;
  const int n   = dn / DD;
  const int tid = threadIdx.x;

  const size_t base = ((size_t)((n*DD + d)*FHH) * FWW + w) * CDIM;

#ifdef HAS_TDM
  if (tid == 0) {
    uint64_t ga = (uint64_t)(uintptr_t)(feats + base);
    uint32_t la = (uint32_t)(uintptr_t)(&tile[0]);
    u32x4 g0;
    g0.x = 1u;                                   // count=1 valid user D#
    g0.y = la;                                   // lds_addr (bytes)
    g0.z = (uint32_t)ga;                         // global_addr[31:0]
    g0.w = (uint32_t)((ga >> 32) & 0x1FFFFFFu)   // global_addr[56:32]
         | (2u << 30);                           // type = 2 ("image")
    const uint32_t dim0 = CDIM, dim1 = FHH, t0 = CDIM, t1 = FHH;
    const uint64_t s0 = (uint64_t)(FWW * CDIM);  // dim0 stride in elements
    i32x8 g1;
    g1[0] = (int)(2u << 16);                     // data_size code 2 = 4 bytes
    g1[1] = (int)(dim0 << 16);                   // tensor_dim0[15:0]
    g1[2] = (int)((dim0 >> 16) | (dim1 << 16));  // dim0[31:16] | dim1[15:0]
    g1[3] = (int)((dim1 >> 16) | (t0 << 16));    // dim1[31:16] | tile_dim0
    g1[4] = (int)t1;                             // tile_dim1 | tile_dim2=0
    g1[5] = (int)(uint32_t)(s0 & 0xFFFFFFFFu);   // dim0_stride[31:0]
    g1[6] = (int)(uint32_t)((s0 >> 32) & 0xFFFFu); // dim0_stride[47:32]
    g1[7] = 0;
    i32x4 gz = {0, 0, 0, 0};
#if __clang_major__ >= 23
    i32x8 gz8 = {0, 0, 0, 0, 0, 0, 0, 0};
    __builtin_amdgcn_tensor_load_to_lds(g0, g1, gz, gz, gz8, 0);
#else
    __builtin_amdgcn_tensor_load_to_lds(g0, g1, gz, gz, 0);
#endif
#ifdef HAS_TWAIT
    __builtin_amdgcn_s_wait_tensorcnt(0);
#else
    asm volatile("s_wait_tensorcnt 0x0" ::: "memory");
#endif
  }
#else
  for (int j = tid; j < FHH * CDIM; j += 96)
    tile[j] = feats[base + (size_t)(j / CDIM) * (FWW * CDIM) + (j % CDIM)];
#endif

  // phase 1: voxel index per image row h (threads 0..31)
  if (tid < FHH) {
    const int h = tid;
    const float* cm = P + 12 + n*24;
    float fx = (float)w * (703.0f / 87.0f);   // linspace(0, IW-1, FW)
    float fy = (float)h * (255.0f / 31.0f);   // linspace(0, IH-1, FH)
    float fz = (float)(d + 1);                // depth bin 1..59
    float ax = fx - cm[9], ay = fy - cm[10], az = fz - cm[11];
    float qx = cm[0]*ax + cm[1]*ay + cm[2]*az;
    float qy = cm[3]*ax + cm[4]*ay + cm[5]*az;
    float qz = cm[6]*ax + cm[7]*ay + cm[8]*az;
    qx *= qz; qy *= qz;                        // un-normalize by depth
    float gx = cm[12]*qx + cm[13]*qy + cm[14]*qz + cm[21];
    float gy = cm[15]*qx + cm[16]*qy + cm[17]*qz + cm[22];
    float gzv = cm[18]*qx + cm[19]*qy + cm[20]*qz + cm[23];
    int ix = (int)((gx  + 54.0f) / 0.3f);      // trunc-toward-zero == astype(int32)
    int iy = (int)((gy  + 54.0f) / 0.3f);
    int iz = (int)((gzv + 10.0f) / 20.0f);
    bool kept = (ix >= 0) & (ix < NX) & (iy >= 0) & (iy < NY) & (iz == 0);
    sidx[h] = kept ? (ix * NY + iy) : -1;
  }
  __syncthreads();
  asm volatile("" ::: "memory");               // keep LDS reads below the fill

  // phase 2: run-length aggregate over h, one thread per channel
  if (tid < CDIM) {
    const int c = tid;
    float acc = 0.0f;
    int cur = sidx[0];
    for (int h = 0; h < FHH; ++h) {
      int id = sidx[h];
      if (id != cur) {
        if (cur >= 0) atomicAdd(&bev[(size_t)c * (NX*NY) + cur], acc);
        acc = 0.0f;
        cur = id;
      }
      if (id >= 0) acc += tile[h * CDIM + c];
    }
    if (cur >= 0) atomicAdd(&bev[(size_t)c * (NX*NY) + cur], acc);
  }
}

// ---------------------------------------------------------------- utilities
__device__ __forceinline__ void inv3(const float* m, float* o) {
  float a=m[0],b=m[1],c=m[2],d=m[3],e=m[4],f=m[5],g=m[6],h=m[7],i=m[8];
  float A =  (e*i - f*h);
  float B = -(d*i - f*g);
  float C =  (d*h - e*g);
  float det = a*A + b*B + c*C;
  float r = 1.0f / det;
  o[0]=A*r;            o[1]=-(b*i - c*h)*r; o[2]= (b*f - c*e)*r;
  o[3]=B*r;            o[4]= (a*i - c*g)*r; o[5]=-(a*f - c*d)*r;
  o[6]=C*r;            o[7]=-(a*h - b*g)*r; o[8]= (a*e - b*d)*r;
}

__device__ __forceinline__ void mul3(const float* a, const float* b, float* o) {
  for (int i = 0; i < 3; ++i)
    for (int j = 0; j < 3; ++j)
      o[i*3+j] = a[i*3+0]*b[0*3+j] + a[i*3+1]*b[1*3+j] + a[i*3+2]*b[2*3+j];
}

// ---------------------------------------------------------------- zero init
// One global_store_b128 per thread; tail handled by thread 0.
__global__ __launch_bounds__(256) void zero_kernel(float* __restrict__ p, int n) {
  int n4 = n >> 2;
  int i = blockIdx.x * blockDim.x + threadIdx.x;
  if (i < n4) {
    float4 z; z.x = 0.0f; z.y = 0.0f; z.z = 0.0f; z.w = 0.0f;
    ((float4*)p)[i] = z;
  }
  if (i == 0) {
    for (int k = n4 << 2; k < n; ++k) p[k] = 0.0f;
  }
}

// ------------------------------------------------------- per-camera matrices
// ws layout (floats):
//  [0..8]   invExRot (row major)         [9..11] ex_tr
//  per cam n at 12 + n*24:
//    [0..8]  invPost   [9..11] post_trans
//    [12..20] M = exRot @ c2lRot @ inv(K)   [21..23] T = exRot@c2l_t + ex_tr
__global__ void prep_kernel(const float* __restrict__ intr,
                            const float* __restrict__ c2l,
                            const float* __restrict__ aug,
                            const float* __restrict__ laug,
                            float* __restrict__ P) {
  if (threadIdx.x != 0 || blockIdx.x != 0) return;
  float exr[9], iexr[9], etr[3];
  for (int i = 0; i < 3; ++i)
    for (int j = 0; j < 3; ++j) exr[i*3+j] = laug[i*4+j];
  inv3(exr, iexr);
  for (int k = 0; k < 9; ++k) P[k] = iexr[k];
  for (int i = 0; i < 3; ++i) { etr[i] = laug[i*4+3]; P[9+i] = etr[i]; }

  for (int n = 0; n < NCAM; ++n) {
    const float* A = aug  + n*16;
    const float* K = intr + n*16;
    const float* C = c2l  + n*16;
    float pr[9], ipr[9], kk[9], ik[9], cr[9], comb[9], M[9];
    for (int i = 0; i < 3; ++i)
      for (int j = 0; j < 3; ++j) {
        pr[i*3+j] = A[i*4+j];
        kk[i*3+j] = K[i*4+j];
        cr[i*3+j] = C[i*4+j];
      }
    inv3(pr, ipr);
    inv3(kk, ik);
    mul3(cr, ik, comb);
    mul3(exr, comb, M);
    float* o = P + 12 + n*24;
    for (int k = 0; k < 9; ++k) o[k] = ipr[k];
    for (int i = 0; i < 3; ++i) o[9+i] = A[i*4+3];
    for (int k = 0; k < 9; ++k) o[12+k] = M[k];
    float ct0 = C[3], ct1 = C[7], ct2 = C[11];
    for (int i = 0; i < 3; ++i)
      o[21+i] = exr[i*3+0]*ct0 + exr[i*3+1]*ct1 + exr[i*3+2]*ct2 + etr[i];
  }
}

// ------------------------------------------------------------ depth scatter
__global__ __launch_bounds__(256) void depth_kernel(
    const float* __restrict__ pts,
    const float* __restrict__ l2i,
    const float* __restrict__ aug,
    const float* __restrict__ P,
    float* __restrict__ depth) {
  int g = blockIdx.x * blockDim.x + threadIdx.x;
  if (g >= PE) return;
  int i = g >> 3;
  int r = g & 7;
  float px = pts[i*5+0], py = pts[i*5+1];
  float f3 = pts[i*5+3], f4 = pts[i*5+4];
  float z  = 0.25f * (float)(r + 1);          // tiled arange(0.25, 2.25, 0.25)
  float x0 = px - P[9], y0 = py - P[10], z0 = z - P[11];
  float cx = P[0]*x0 + P[1]*y0 + P[2]*z0;     // inv(ex_rot) @ xyz
  float cy = P[3]*x0 + P[4]*y0 + P[5]*z0;
  float cz = P[6]*x0 + P[7]*y0 + P[8]*z0;
  for (int n = 0; n < NCAM; ++n) {
    const float* L = l2i + n*16;
    float ux = L[0]*cx + L[1]*cy + L[2]*cz  + L[3];
    float uy = L[4]*cx + L[5]*cy + L[6]*cz  + L[7];
    float uz = L[8]*cx + L[9]*cy + L[10]*cz + L[11];
    float dist = uz;
    float zc = fminf(fmaxf(uz, 1e-5f), 1e5f);
    float vx = ux / zc, vy = uy / zc, vz = zc;
    const float* A = aug + n*16;
    float wx = A[0]*vx + A[1]*vy + A[2]*vz + A[3];
    float wy = A[4]*vx + A[5]*vy + A[6]*vz + A[7];
    if (wx >= 0.0f && wx < (float)IWW && wy >= 0.0f && wy < (float)IHH) {
      int xi = (int)wx, yi = (int)wy;
      float* dp = depth + (size_t)(n*6) * PIX + (size_t)yi * IWW + xi;
      dp[0*PIX] = dist;
      dp[1*PIX] = x0;
      dp[2*PIX] = y0;
      dp[3*PIX] = z0;
      dp[4*PIX] = f3;
      dp[5*PIX] = f4;
    }
  }
}

// ----------------------------------------------------------------- launcher
extern "C" void kernel_launch(void* const* d_in, const int* in_sizes, int n_in,
                              void* d_out, int out_size, void* d_ws, size_t ws_size,
                              hipStream_t stream) {
  const float* cam_feats = (const float*)d_in[0];
  const float* points    = (const float*)d_in[1];
  const float* l2i       = (const float*)d_in[2];
  const float* intr      = (const float*)d_in[3];
  const float* c2l       = (const float*)d_in[4];
  const float* aug       = (const float*)d_in[5];
  const float* laug      = (const float*)d_in[6];

  float* out   = (float*)d_out;
  float* bev   = out;                 // (80, 360, 360)
  float* depth = out + BEV_ELEMS;     // (6, 6, 256, 704)
  float* P     = (float*)d_ws;        // 156 floats of precomputed matrices

  int nz4 = (out_size >> 2) + 1;
  zero_kernel<<<(nz4 + 255) / 256, 256, 0, stream>>>(out, out_size);
  prep_kernel<<<1, 1, 0, stream>>>(intr, c2l, aug, laug, P);
  bev_pool_kernel<<<NCAM * DD * FWW, 96, 0, stream>>>(cam_feats, P, bev);
  depth_kernel<<<(PE + 255) / 256, 256, 0, stream>>>(points, l2i, aug, P, depth);
}